// cross_set_score_66709432042165
// MI455X (gfx1250) — compile-verified
//
#include <hip/hip_runtime.h>

typedef float v2f __attribute__((ext_vector_type(2)));
typedef float v8f __attribute__((ext_vector_type(8)));

#define NSET 64
#define NITEM 64
#define DIN 256
#define DOUT 512   // NUM_HEADS * HEAD_SIZE

// ---------------------------------------------------------------------------
// Kernel 1: X̄[j,i,f] = sum_a x[(j*64+a), i, f]
// x: [4096, 64, 256] f32 (268 MB)  ->  xbar: [64*64, 256] f32 (4 MB)
// grid = (16 i-groups, 64 j), block = 256 threads.
// Each thread owns one float4 of f; block covers i in [4b,4b+4) x f in [0,256)
// so each a-iteration is one contiguous 4 KB read per block. Pure streaming.
// ---------------------------------------------------------------------------
__global__ void reduce_a_kernel(const float* __restrict__ x,
                                float* __restrict__ xbar) {
    const int j  = blockIdx.y;
    const int i  = blockIdx.x * 4 + (threadIdx.x >> 6);
    const int f4 = threadIdx.x & 63;           // float4 index within row

    const float4* xp = reinterpret_cast<const float4*>(x);
    // float index: ((j*64 + a)*64 + i)*256 + 4*f4 ; in float4 units:
    size_t base = (((size_t)j * 64) * 64 + (size_t)i) * 64 + f4;  // a = 0
    const size_t astride = (size_t)64 * 256 / 4;                  // 4096 float4 per a

    float4 acc = make_float4(0.f, 0.f, 0.f, 0.f);
#pragma unroll 8
    for (int a = 0; a < NITEM; ++a) {
        float4 v = xp[base + (size_t)a * astride];
        acc.x += v.x; acc.y += v.y; acc.z += v.z; acc.w += v.w;
    }
    reinterpret_cast<float4*>(xbar)[(size_t)(j * 64 + i) * 64 + f4] = acc;
}

// ---------------------------------------------------------------------------
// Kernel 2: g[4096,512] = xbar[4096,256] @ W[256,512] via V_WMMA_F32_16X16X4_F32
// One wave32 per 16x16 output tile; 256 tiles (M) x 32 tiles (N) = 8192 tiles.
// 8 waves per block -> 1024 blocks. K looped 0..255 step 4 -> 64 WMMAs/wave.
//
// ISA VGPR layouts (CDNA5 ISA):
//  A 16x4 : lanes 0-15 -> M=lane, {v0,v1}={K0,K1}; lanes 16-31 -> {K2,K3}
//  B 4x16 : v0: lanes0-15=row K0, lanes16-31=row K2 ; v1: rows K1/K3
//  C/D    : vr: lanes0-15 -> M=r,N=lane ; lanes16-31 -> M=r+8,N=lane-16
// ---------------------------------------------------------------------------
__global__ void gemm_wmma_kernel(const float* __restrict__ xbar,
                                 const float* __restrict__ W,
                                 float* __restrict__ g) {
    const int wave = threadIdx.x >> 5;
    const int lane = threadIdx.x & 31;
    const int tile = blockIdx.x * 8 + wave;
    const int m0 = (tile >> 5) << 4;     // 0..4080
    const int n0 = (tile & 31) << 4;     // 0..496
    const int hsel = lane >> 4;          // 0: K pair {0,1}, 1: K pair {2,3}
    const int l    = lane & 15;
    const int koff = hsel * 2;

    const float* arow = xbar + (size_t)(m0 + l) * DIN + koff;       // A: M=l row
    const float* bcol = W + (size_t)koff * DOUT + n0 + l;           // B: N=l col

    v8f c = {0.f, 0.f, 0.f, 0.f, 0.f, 0.f, 0.f, 0.f};
#pragma unroll 4
    for (int kk = 0; kk < DIN; kk += 4) {
        v2f a = *reinterpret_cast<const v2f*>(arow + kk);  // K=kk+koff, kk+koff+1
        v2f b;
        b.x = bcol[(size_t)kk * DOUT];                     // row K=kk+koff
        b.y = bcol[(size_t)kk * DOUT + DOUT];              // row K=kk+koff+1
        c = __builtin_amdgcn_wmma_f32_16x16x4_f32(
            /*neg_a=*/false, a, /*neg_b=*/false, b,
            /*c_mod=*/(short)0, c, /*reuse_a=*/false, /*reuse_b=*/false);
    }

    const int row0 = m0 + hsel * 8;
    const int col  = n0 + l;
#pragma unroll
    for (int r = 0; r < 8; ++r) {
        g[(size_t)(row0 + r) * DOUT + col] = c[r];
    }
}

// ---------------------------------------------------------------------------
// Kernel 3: out[j,i] = (1/32768) * sum_e g[j*64+i, e] * g[i*64+j, e] * wc[e>>6]
// One wave32 per (j,i); 8 waves/block -> 512 blocks. shfl_xor wave reduction.
// ---------------------------------------------------------------------------
__global__ void score_kernel(const float* __restrict__ g,
                             const float* __restrict__ wconv,
                             float* __restrict__ out) {
    const int wave = threadIdx.x >> 5;
    const int lane = threadIdx.x & 31;
    const int p = blockIdx.x * 8 + wave;     // 0..4095 == j*64 + i
    const int j = p >> 6;
    const int i = p & 63;

    const float* g1 = g + (size_t)p * DOUT;
    const float* g2 = g + (size_t)(i * 64 + j) * DOUT;

    float acc = 0.f;
#pragma unroll
    for (int e = lane; e < DOUT; e += 32) {
        acc += g1[e] * g2[e] * wconv[e >> 6];
    }
#pragma unroll
    for (int off = 16; off > 0; off >>= 1) {
        acc += __shfl_xor(acc, off, 32);
    }
    if (lane == 0) {
        out[p] = acc * (1.0f / 32768.0f);    // / sqrt(64) / 64 / 64
    }
}

// ---------------------------------------------------------------------------
extern "C" void kernel_launch(void* const* d_in, const int* in_sizes, int n_in,
                              void* d_out, int out_size, void* d_ws, size_t ws_size,
                              hipStream_t stream) {
    const float* x     = (const float*)d_in[0];   // [4096, 64, 256]
    const float* W     = (const float*)d_in[1];   // [256, 512]
    const float* wconv = (const float*)d_in[2];   // [8]
    // d_in[3] = nItem (==64), baked into the kernels.

    float* xbar = (float*)d_ws;                   // 4096*256 f32 = 4 MB
    float* g    = xbar + (size_t)NSET * NSET * DIN; // 4096*512 f32 = 8 MB
    float* out  = (float*)d_out;                  // [64, 64]

    reduce_a_kernel<<<dim3(16, 64), 256, 0, stream>>>(x, xbar);
    gemm_wmma_kernel<<<1024, 256, 0, stream>>>(xbar, W, g);
    score_kernel<<<512, 256, 0, stream>>>(g, wconv, out);
}